// MyModel_16999480557714
// MI455X (gfx1250) — compile-verified
//
#include <hip/hip_runtime.h>
#include <math.h>

typedef float v2f __attribute__((ext_vector_type(2)));
typedef float v8f __attribute__((ext_vector_type(8)));

struct C2 { float x, y; };
// a * (c + i*s)
__device__ __forceinline__ C2 cmulp(C2 a, float c, float s) {
    C2 r; r.x = a.x * c - a.y * s; r.y = a.x * s + a.y * c; return r;
}

#define WAVES_PER_BLOCK 4
#define KDIM 3072

__global__ __launch_bounds__(128) void qfused_kernel(
    const float* __restrict__ x,   // (16384, 3072) flattened
    const float* __restrict__ W1,  // (8, 3072)
    const float* __restrict__ b1,  // (8,)
    const float* __restrict__ q0,  // (3,)
    const float* __restrict__ q1,  // (3,)
    const float* __restrict__ Wc,  // (10, 8)
    const float* __restrict__ bc,  // (10,)
    float* __restrict__ out)       // (16384, 10)
{
    __shared__ float lds[WAVES_PER_BLOCK][16][8];

    const int lane = threadIdx.x & 31;
    const int wave = threadIdx.x >> 5;
    const int tile = blockIdx.x * WAVES_PER_BLOCK + wave;
    const int rowBase = tile * 16;

    const int m    = lane & 15;      // A row / B col / C col within tile
    const int kh   = lane >> 4;      // which half of the wave
    const int koff = kh * 2;         // K sub-offset for this lane (A & B frags)
    const int n    = m;              // output feature column (0..15, only 0..7 real)

    // NOTE: no pad-masking needed. For n in [8,15] we load W1 row (n&7), so
    // accumulator columns 8..15 simply duplicate columns 0..7; they are never
    // read (LDS store below is guarded by n < 8). This keeps the inner loop
    // at exactly 2 x global_load_b64 + 1 x v_wmma per K-step.
    const float* aPtr = x  + (size_t)(rowBase + m) * KDIM + koff;
    const float* bPtr = W1 + (size_t)(n & 7)       * KDIM + koff;

    // ---- main GEMM: h = x @ W1.T, accumulated via V_WMMA_F32_16X16X4_F32 ----
    v8f acc = {};
#pragma unroll 8
    for (int k = 0; k < KDIM; k += 4) {
        v2f a = *(const v2f*)(aPtr + k);
        v2f b = *(const v2f*)(bPtr + k);
        acc = __builtin_amdgcn_wmma_f32_16x16x4_f32(
            /*neg_a=*/false, a, /*neg_b=*/false, b,
            /*c_mod=*/(short)0, acc, /*reuse_a=*/false, /*reuse_b=*/false);
    }

    // ---- per-launch-constant rotation angles (tiny loads, L2-resident) ----
    const float cg0 = cosf(0.5f * q0[0]), sg0 = sinf(0.5f * q0[0]); // RZ q0[0]
    const float cb0 = cosf(0.5f * q0[1]), sb0 = sinf(0.5f * q0[1]); // RY q0[1]
    const float cg2 = cosf(0.5f * q0[2]), sg2 = sinf(0.5f * q0[2]); // RZ q0[2]
    const float ch0 = cosf(0.5f * q1[0]), sh0 = sinf(0.5f * q1[0]); // RZ q1[0]
    const float cb1 = cosf(0.5f * q1[1]), sb1 = sinf(0.5f * q1[1]); // RY q1[1]
    // final RZ(q1[2]) is a pure phase: |p0|^2,|p1|^2 unchanged -> dropped.
    const float bias = b1[n & 7];

    // ---- quantum circuit closed-form on each h element of this lane's column ----
#pragma unroll
    for (int r = 0; r < 8; ++r) {
        float hv = acc[r] + bias;
        float t = 0.5f * hv;
        float s, c;
        __sincosf(t, &s, &c);

        C2 p0; p0.x = c; p0.y = 0.0f;   // after RY(ang) from |0>
        C2 p1; p1.x = s; p1.y = 0.0f;

        // RZ(q0[0]):  p0 *= e^{-i t/2},  p1 *= e^{+i t/2}
        p0 = cmulp(p0, cg0, -sg0);  p1 = cmulp(p1, cg0, sg0);
        // RY(q0[1])
        { C2 n0, n1;
          n0.x = cb0*p0.x - sb0*p1.x;  n0.y = cb0*p0.y - sb0*p1.y;
          n1.x = sb0*p0.x + cb0*p1.x;  n1.y = sb0*p0.y + cb0*p1.y;
          p0 = n0; p1 = n1; }
        // RZ(q0[2])
        p0 = cmulp(p0, cg2, -sg2);  p1 = cmulp(p1, cg2, sg2);
        // RY(ang) again
        { C2 n0, n1;
          n0.x = c*p0.x - s*p1.x;  n0.y = c*p0.y - s*p1.y;
          n1.x = s*p0.x + c*p1.x;  n1.y = s*p0.y + c*p1.y;
          p0 = n0; p1 = n1; }
        // RZ(q1[0])
        p0 = cmulp(p0, ch0, -sh0);  p1 = cmulp(p1, ch0, sh0);
        // RY(q1[1])
        { C2 n0, n1;
          n0.x = cb1*p0.x - sb1*p1.x;  n0.y = cb1*p0.y - sb1*p1.y;
          n1.x = sb1*p0.x + cb1*p1.x;  n1.y = sb1*p0.y + cb1*p1.y;
          p0 = n0; p1 = n1; }

        float z  = (p0.x*p0.x + p0.y*p0.y) - (p1.x*p1.x + p1.y*p1.y);
        float qv = 0.5f * (z + 1.0f);

        int row = r + kh * 8;  // C/D layout: reg r -> M = r + 8*(lane>>4)
        if (n < 8) lds[wave][row][n] = qv;
    }

    __syncthreads();

    // ---- classifier: out = sigmoid(fuzz @ Wc.T + bc), 16 rows x 10 outputs ----
    for (int t10 = lane; t10 < 160; t10 += 32) {
        int row = t10 / 10;
        int j   = t10 % 10;
        float v = bc[j];
#pragma unroll
        for (int nn = 0; nn < 8; ++nn)
            v += lds[wave][row][nn] * Wc[j * 8 + nn];
        out[(size_t)(rowBase + row) * 10 + j] = 1.0f / (1.0f + expf(-v));
    }
}

extern "C" void kernel_launch(void* const* d_in, const int* in_sizes, int n_in,
                              void* d_out, int out_size, void* d_ws, size_t ws_size,
                              hipStream_t stream) {
    (void)in_sizes; (void)n_in; (void)out_size; (void)d_ws; (void)ws_size;
    const float* x  = (const float*)d_in[0];
    const float* W1 = (const float*)d_in[1];
    const float* b1 = (const float*)d_in[2];
    const float* q0 = (const float*)d_in[3];
    const float* q1 = (const float*)d_in[4];
    const float* Wc = (const float*)d_in[5];
    const float* bc = (const float*)d_in[6];
    float* out = (float*)d_out;

    // 16384 rows / 16 per wave = 1024 tiles; 4 waves (128 threads) per block.
    dim3 grid(1024 / WAVES_PER_BLOCK);
    dim3 block(128);
    qfused_kernel<<<grid, block, 0, stream>>>(x, W1, b1, q0, q1, Wc, bc, out);
}